// AggressiveLoss_37649683317562
// MI455X (gfx1250) — compile-verified
//
#include <hip/hip_runtime.h>
#include <math.h>

// ---------------------------------------------------------------------------
// AggressiveLoss on MI455X (gfx1250, wave32).
// Memory-bound streaming reduction: ~110.6 MB read -> ~4.7us at 23.3 TB/s.
// Hot loop: coalesced float4 (B128) loads, register-resident channel math,
// LDS tree reduction, one f64 + four f32 global atomics per block.
// Finalize: v_wmma_f32_16x16x4_f32 sums the loss components (f32 precision).
// ---------------------------------------------------------------------------

typedef __attribute__((ext_vector_type(2))) float v2f;
typedef __attribute__((ext_vector_type(8))) float v8f;

#define NC      10
#define HW      900      // 30*30 pixels per image
#define GROUPS  225      // HW / 4 pixels-per-thread
#define BLOCK   256

struct WS {
    double ce_sum;          // sum over all pixels of ce * weight
    float  exact_cnt;       // # batches with all pixels pred==tgt
    float  copy_cnt;        // # batches with should_not_copy && did_copy
    float  transform_sum;   // sum over batches of (changed - target_changed)^2
    float  missing_cnt;     // total missing colors across batches
};

__global__ void init_ws_kernel(WS* __restrict__ ws) {
    if (threadIdx.x == 0) {
        ws->ce_sum        = 0.0;
        ws->exact_cnt     = 0.0f;
        ws->copy_cnt      = 0.0f;
        ws->transform_sum = 0.0f;
        ws->missing_cnt   = 0.0f;
    }
}

__global__ __launch_bounds__(BLOCK) void loss_main_kernel(
        const float* __restrict__ pred,
        const float* __restrict__ tgt,
        const float* __restrict__ inp,
        WS* __restrict__ ws) {
    __shared__ float s_ce[BLOCK];
    __shared__ int   s_mis[BLOCK];   // pred_idx != tgt_idx count
    __shared__ int   s_pni[BLOCK];   // pred_idx != inp_idx count
    __shared__ int   s_tni[BLOCK];   // tgt_idx  != inp_idx count
    __shared__ int   s_pm[BLOCK];    // pred color bitmask (OR)
    __shared__ int   s_tm[BLOCK];    // tgt  color bitmask (OR)

    const int b   = blockIdx.x;
    const int tid = threadIdx.x;

    float ce_acc = 0.0f;
    int mis = 0, pni = 0, tni = 0, pm = 0, tm = 0;

    if (tid < GROUPS) {
        const size_t base = (size_t)b * (NC * HW) + (size_t)tid * 4;

        float4 buf[NC];
        const float* bp = reinterpret_cast<const float*>(buf);
        int ti[4], ii[4];

        // ---- target: per-pixel argmax (first occurrence) ----
        #pragma unroll
        for (int c = 0; c < NC; ++c)
            buf[c] = *reinterpret_cast<const float4*>(tgt + base + (size_t)c * HW);
        #pragma unroll
        for (int j = 0; j < 4; ++j) {
            float m = bp[j]; int idx = 0;
            #pragma unroll
            for (int c = 1; c < NC; ++c) {
                float v = bp[c * 4 + j];
                if (v > m) { m = v; idx = c; }
            }
            ti[j] = idx;
        }

        // ---- input_grid: per-pixel argmax ----
        #pragma unroll
        for (int c = 0; c < NC; ++c)
            buf[c] = *reinterpret_cast<const float4*>(inp + base + (size_t)c * HW);
        #pragma unroll
        for (int j = 0; j < 4; ++j) {
            float m = bp[j]; int idx = 0;
            #pragma unroll
            for (int c = 1; c < NC; ++c) {
                float v = bp[c * 4 + j];
                if (v > m) { m = v; idx = c; }
            }
            ii[j] = idx;
        }

        // ---- pred: argmax + logsumexp CE at target channel ----
        #pragma unroll
        for (int c = 0; c < NC; ++c)
            buf[c] = *reinterpret_cast<const float4*>(pred + base + (size_t)c * HW);
        #pragma unroll
        for (int j = 0; j < 4; ++j) {
            float m = bp[j]; int pi = 0;
            #pragma unroll
            for (int c = 1; c < NC; ++c) {
                float v = bp[c * 4 + j];
                if (v > m) { m = v; pi = c; }
            }
            const int tj = ti[j];
            float es = 0.0f;
            float v_at_t = 0.0f;
            #pragma unroll
            for (int c = 0; c < NC; ++c) {
                float v = bp[c * 4 + j];
                es += expf(v - m);
                v_at_t = (c == tj) ? v : v_at_t;
            }
            const float lse = m + logf(es);
            const float ce  = lse - v_at_t;           // -log_softmax[tgt]
            const int wrong = (pi != tj) ? 1 : 0;
            ce_acc += ce * (wrong ? 5.0f : 1.0f);
            mis += wrong;
            pni += (pi != ii[j]) ? 1 : 0;
            tni += (tj != ii[j]) ? 1 : 0;
            pm  |= (1 << pi);
            tm  |= (1 << tj);
        }
    }

    s_ce[tid]  = ce_acc;
    s_mis[tid] = mis;
    s_pni[tid] = pni;
    s_tni[tid] = tni;
    s_pm[tid]  = pm;
    s_tm[tid]  = tm;
    __syncthreads();

    #pragma unroll
    for (int s = BLOCK / 2; s > 0; s >>= 1) {
        if (tid < s) {
            s_ce[tid]  += s_ce[tid + s];
            s_mis[tid] += s_mis[tid + s];
            s_pni[tid] += s_pni[tid + s];
            s_tni[tid] += s_tni[tid + s];
            s_pm[tid]  |= s_pm[tid + s];
            s_tm[tid]  |= s_tm[tid + s];
        }
        __syncthreads();
    }

    if (tid == 0) {
        atomicAdd(&ws->ce_sum, (double)s_ce[0]);
        if (s_mis[0] == 0) atomicAdd(&ws->exact_cnt, 1.0f);
        if (s_tni[0] > 0 && s_pni[0] == 0) atomicAdd(&ws->copy_cnt, 1.0f);
        const float inv_hw  = 1.0f / (float)HW;
        const float changed = (float)s_pni[0] * inv_hw;
        const float tchg    = (float)s_tni[0] * inv_hw;
        const float d       = changed - tchg;
        atomicAdd(&ws->transform_sum, d * d);
        const int missing = __popc(s_tm[0] & ~s_pm[0]);
        if (missing) atomicAdd(&ws->missing_cnt, (float)missing);
    }
}

__global__ void finalize_kernel(const WS* __restrict__ ws,
                                float* __restrict__ out, int B) {
    const float inv_b      = 1.0f / (float)B;
    const float ce_loss    = (float)(ws->ce_sum / ((double)B * (double)HW));
    const float exact_rate = ws->exact_cnt * inv_b;
    const float copy_pen   = 5.0f * ws->copy_cnt * inv_b;
    const float transform  = ws->transform_sum * inv_b;
    const float color_pen  = 0.1f * ws->missing_cnt;

    // Sum the loss terms with V_WMMA_F32_16X16X4_F32 (full f32 precision).
    // A (16x4) layout: lanes 0-15 supply K=0,1; lanes 16-31 supply K=2,3.
    // With B = ones, every D element = row-sum of A = total loss.
    const int lane = threadIdx.x;
    v2f a, bb;
    a.x = (lane < 16) ? ce_loss     : copy_pen;
    a.y = (lane < 16) ? -exact_rate : (transform + color_pen);
    bb.x = 1.0f;
    bb.y = 1.0f;
    v8f c = {};
    v8f d = __builtin_amdgcn_wmma_f32_16x16x4_f32(
        /*neg_a=*/false, a, /*neg_b=*/false, bb,
        /*c_mod=*/(short)0, c, /*reuse_a=*/false, /*reuse_b=*/false);

    float total = d[0];
    if (total != total)        total = 2.0f;   // NaN guard
    else if (total > 100.0f)   total = 10.0f;

    if (lane == 0) {
        out[0] = total;
        out[1] = ce_loss;
        out[2] = copy_pen;
        out[3] = exact_rate;      // -exact_bonus
        out[4] = ws->exact_cnt;   // sum(exact_matches)
        out[5] = transform;
    }
}

extern "C" void kernel_launch(void* const* d_in, const int* in_sizes, int n_in,
                              void* d_out, int out_size, void* d_ws, size_t ws_size,
                              hipStream_t stream) {
    (void)n_in; (void)out_size; (void)ws_size;
    const float* pred = (const float*)d_in[0];
    const float* tgt  = (const float*)d_in[1];
    const float* inp  = (const float*)d_in[2];
    float* out = (float*)d_out;
    WS* ws = (WS*)d_ws;

    const int B = in_sizes[0] / (NC * HW);

    hipLaunchKernelGGL(init_ws_kernel, dim3(1), dim3(32), 0, stream, ws);
    hipLaunchKernelGGL(loss_main_kernel, dim3(B), dim3(BLOCK), 0, stream,
                       pred, tgt, inp, ws);
    hipLaunchKernelGGL(finalize_kernel, dim3(1), dim3(32), 0, stream, ws, out, B);
}